// SSIM_67388036874334
// MI455X (gfx1250) — compile-verified
//
#include <hip/hip_runtime.h>
#include <math.h>

// SSIM on [16,3,512,512] fp32 via separable 11x11 Gaussian conv mapped onto
// V_WMMA_F32_16X16X4_F32 banded matmuls. One wave32 per 16x16 output tile.

typedef __attribute__((ext_vector_type(2))) float v2f;
typedef __attribute__((ext_vector_type(8))) float v8f;

#define IMG_H 512
#define IMG_W 512
#define NIMG 48                      // B*C = 16*3
#define TILES_X 32
#define TILES_PER_IMG (TILES_X * TILES_X)          // 1024
#define TOTAL_WAVES (NIMG * TILES_PER_IMG)         // 49152
#define WAVES_PER_BLOCK 4
#define HALO_ROWS 28                 // 16 + 2*5 halo, padded to 7 k-steps of 4
#define HS 33                        // LDS row stride (floats), bank-conflict pad
#define WAVE_LDS (2 * HALO_ROWS * HS + 16 * HS)    // 2376 floats per wave
#define N_ELEMS 12582912.0f          // 16*3*512*512

__device__ __forceinline__ float clipv(float x) {
    return fminf(fmaxf(x, 1e-6f), 1.0f - 1e-6f);
}

// 1D gaussian coefficient g[i] (sigma=1.5, 11 taps, normalized), 0 out of range
__device__ __forceinline__ float gval(int i, float inv) {
    if (i < 0 || i > 10) return 0.0f;
    float d = (float)(i - 5);
    return expf(-d * d * (1.0f / 4.5f)) * inv;
}

__global__ __launch_bounds__(128) void ssim_tile_kernel(
    const float* __restrict__ img1, const float* __restrict__ img2,
    float* __restrict__ wavesums)
{
    __shared__ float lds[WAVES_PER_BLOCK * WAVE_LDS];

    const int lane = threadIdx.x & 31;
    const int wave = threadIdx.x >> 5;
    const int wgid = blockIdx.x * WAVES_PER_BLOCK + wave;   // global tile id

    float* h1  = &lds[wave * WAVE_LDS];          // clipped img1 halo 28 x 32
    float* h2  = h1 + HALO_ROWS * HS;            // clipped img2 halo 28 x 32
    float* scr = h2 + HALO_ROWS * HS;            // 16 x 32 transpose scratch

    const int img = wgid / TILES_PER_IMG;
    const int rem = wgid % TILES_PER_IMG;
    const int ty  = rem / TILES_X;
    const int tx  = rem % TILES_X;
    const int gy0 = ty * 16 - 5;
    const int gx0 = tx * 16 - 5;
    const int base = img * (IMG_H * IMG_W);

    // ---- gaussian normalization (runtime, matches reference window) ----
    float gsum = 0.0f;
    #pragma unroll
    for (int i = 0; i < 11; ++i) {
        float d = (float)(i - 5);
        gsum += expf(-d * d * (1.0f / 4.5f));
    }
    const float inv = 1.0f / gsum;

    const int m16   = lane & 15;            // row (A/D) or column (B) index
    const int khalf = (lane >> 4) << 1;     // 0 or 2: lane-half K offset

    // Band fragments: Gv A-frag elem (m,k)=g[k-m]; Gh B-frag elem (k,n)=g[k-n].
    // With n == m16 these are identical -> one fragment set serves both passes.
    v2f gf[7];
    #pragma unroll
    for (int kk = 0; kk < 7; ++kk) {
        int k0 = 4 * kk + khalf;
        v2f t;
        t.x = gval(k0     - m16, inv);
        t.y = gval(k0 + 1 - m16, inv);
        gf[kk] = t;
    }

    // ---- stage clipped, zero-padded halo into LDS (lane = column) ----
    #pragma unroll 4
    for (int r = 0; r < HALO_ROWS; ++r) {
        int gy = gy0 + r;
        int gx = gx0 + lane;
        float v1 = 0.0f, v2 = 0.0f;
        if (gy >= 0 && gy < IMG_H && gx >= 0 && gx < IMG_W) {
            int idx = base + gy * IMG_W + gx;
            v1 = clipv(img1[idx]);
            v2 = clipv(img2[idx]);
        }
        h1[r * HS + lane] = v1;
        h2[r * HS + lane] = v2;
    }

    // ---- 5 separable convolutions: f0=x, f1=y, f2=x*x, f3=y*y, f4=x*y ----
    v8f res[5];
    #pragma unroll
    for (int f = 0; f < 5; ++f) {
        // vertical pass: two 16-column chunks -> scratch (16 x 26 valid)
        #pragma unroll
        for (int cc = 0; cc < 2; ++cc) {
            v8f acc = {0.f, 0.f, 0.f, 0.f, 0.f, 0.f, 0.f, 0.f};
            int col = cc * 16 + m16;
            #pragma unroll
            for (int kk = 0; kk < 7; ++kk) {
                int r0 = 4 * kk + khalf;
                float x1a = h1[ r0      * HS + col];
                float x1b = h1[(r0 + 1) * HS + col];
                float x2a = h2[ r0      * HS + col];
                float x2b = h2[(r0 + 1) * HS + col];
                v2f bv;
                if (f == 0)      { bv.x = x1a;       bv.y = x1b;       }
                else if (f == 1) { bv.x = x2a;       bv.y = x2b;       }
                else if (f == 2) { bv.x = x1a * x1a; bv.y = x1b * x1b; }
                else if (f == 3) { bv.x = x2a * x2a; bv.y = x2b * x2b; }
                else             { bv.x = x1a * x2a; bv.y = x1b * x2b; }
                acc = __builtin_amdgcn_wmma_f32_16x16x4_f32(
                    false, gf[kk], false, bv, (short)0, acc, false, false);
            }
            // D layout: VGPR r holds M = r + 8*(lane>>4), N = lane&15
            int mrow = (lane >> 4) * 8;
            #pragma unroll
            for (int r = 0; r < 8; ++r)
                scr[(mrow + r) * HS + col] = acc[r];
        }
        // horizontal pass: A-frag gathered from scratch (row m16, cols k0..k0+1)
        v8f acc2 = {0.f, 0.f, 0.f, 0.f, 0.f, 0.f, 0.f, 0.f};
        #pragma unroll
        for (int kk = 0; kk < 7; ++kk) {
            int k0 = 4 * kk + khalf;
            v2f av;
            av.x = scr[m16 * HS + k0];
            av.y = scr[m16 * HS + k0 + 1];
            acc2 = __builtin_amdgcn_wmma_f32_16x16x4_f32(
                false, av, false, gf[kk], (short)0, acc2, false, false);
        }
        res[f] = acc2;
    }

    // ---- SSIM map on the 8 per-lane accumulator elements + wave reduce ----
    const float c1 = 1e-4f;   // 0.01^2
    const float c2 = 9e-4f;   // 0.03^2
    float sum = 0.0f;
    #pragma unroll
    for (int r = 0; r < 8; ++r) {
        float mu1 = res[0][r], mu2 = res[1][r];
        float e11 = res[2][r], e22 = res[3][r], e12 = res[4][r];
        float mu1s = mu1 * mu1, mu2s = mu2 * mu2, m12 = mu1 * mu2;
        float s1  = e11 - mu1s;
        float s2  = e22 - mu2s;
        float s12 = e12 - m12;
        float num = (2.0f * m12 + c1) * (2.0f * s12 + c2);
        float den = (mu1s + mu2s + c1) * (s1 + s2 + c2);
        sum += num / den;
    }
    #pragma unroll
    for (int off = 16; off > 0; off >>= 1)
        sum += __shfl_xor(sum, off, 32);
    if (lane == 0)
        wavesums[wgid] = sum;
}

__global__ __launch_bounds__(256) void ssim_finalize_kernel(
    const float* __restrict__ wavesums, float* __restrict__ out)
{
    __shared__ float sm[256];
    float a = 0.0f;
    for (int i = threadIdx.x; i < TOTAL_WAVES; i += 256)
        a += wavesums[i];
    sm[threadIdx.x] = a;
    __syncthreads();
    for (int s = 128; s > 0; s >>= 1) {
        if (threadIdx.x < s) sm[threadIdx.x] += sm[threadIdx.x + s];
        __syncthreads();
    }
    if (threadIdx.x == 0)
        out[0] = 1.0f - sm[0] * (1.0f / N_ELEMS);
}

extern "C" void kernel_launch(void* const* d_in, const int* in_sizes, int n_in,
                              void* d_out, int out_size, void* d_ws, size_t ws_size,
                              hipStream_t stream) {
    (void)in_sizes; (void)n_in; (void)out_size; (void)ws_size;
    const float* img1 = (const float*)d_in[0];
    const float* img2 = (const float*)d_in[1];
    // d_in[2] (window) is reproduced analytically in-kernel (sigma = 1.5)
    float* wavesums = (float*)d_ws;            // TOTAL_WAVES floats (192 KB)
    float* out = (float*)d_out;

    ssim_tile_kernel<<<TOTAL_WAVES / WAVES_PER_BLOCK, 128, 0, stream>>>(
        img1, img2, wavesums);
    ssim_finalize_kernel<<<1, 256, 0, stream>>>(wavesums, out);
}